// PostProcess_56547539419222
// MI455X (gfx1250) — compile-verified
//
#include <hip/hip_runtime.h>
#include <stdint.h>

#define CONF_THRES 0.25f
#define IOU_THRES  0.45f
#define DET        300
#define K_TOP      1024
#define MAX_COORD  4.0f
#define BATCH      16
#define NROWS      25200
#define NCLS       80
#define ROWLEN     85
#define NBINS      2048
#define CAP        8192

typedef __attribute__((ext_vector_type(2))) float v2f;
typedef __attribute__((ext_vector_type(8))) float v8f;

extern __shared__ unsigned char dyn_smem[];

__device__ __forceinline__ int bin_of(float s) {
  int b = (int)((s - CONF_THRES) * ((float)NBINS / 0.75f));
  if (b < 0) b = 0;
  if (b > NBINS - 1) b = NBINS - 1;
  return b;
}

// ---------------------------------------------------------------- zero scratch
__global__ void zero_kernel(uint32_t* p, int n) {
  int i = blockIdx.x * blockDim.x + threadIdx.x;
  if (i < n) p[i] = 0u;
}

// ------------------------------------------------- pass 1: score histogram (HBM)
__global__ void hist_kernel(const float* __restrict__ pred, int* __restrict__ hist) {
  int b = blockIdx.y;
  int gid = blockIdx.x * blockDim.x + threadIdx.x;
  if (gid >= NROWS * NCLS) return;
  int n = gid / NCLS;
  int c = gid - n * NCLS;
  const float* row = pred + ((size_t)b * NROWS + n) * ROWLEN;
  __builtin_prefetch(row + 64 * ROWLEN, 0, 1);   // global_prefetch_b8
  float s = row[4] * row[5 + c];
  if (s > CONF_THRES) atomicAdd(&hist[b * NBINS + bin_of(s)], 1);
}

// ---------------------------------------- find cutoff bin for exact top-1024
__global__ void scan_kernel(const int* __restrict__ hist, int* __restrict__ cutbin) {
  int b = blockIdx.x;
  if (threadIdx.x != 0) return;
  int acc = 0, cut = 0;
  for (int i = NBINS - 1; i >= 0; --i) {
    acc += hist[b * NBINS + i];
    if (acc >= K_TOP) { cut = i; break; }
  }
  cutbin[b] = cut;
}

// -------------------------------- pass 2: compact candidates (mostly L2 hits)
__global__ void compact_kernel(const float* __restrict__ pred,
                               const int* __restrict__ cutbin,
                               unsigned long long* __restrict__ cand,
                               int* __restrict__ counter) {
  int b = blockIdx.y;
  int gid = blockIdx.x * blockDim.x + threadIdx.x;
  if (gid >= NROWS * NCLS) return;
  int n = gid / NCLS;
  int c = gid - n * NCLS;
  const float* row = pred + ((size_t)b * NROWS + n) * ROWLEN;
  __builtin_prefetch(row + 64 * ROWLEN, 0, 1);
  float s = row[4] * row[5 + c];
  if (s > CONF_THRES && bin_of(s) >= cutbin[b]) {
    // key: score-desc, index-asc (matches lax.top_k tie order)
    unsigned long long key =
        ((unsigned long long)__float_as_uint(s) << 32) | (unsigned)(~(unsigned)gid);
    int pos = atomicAdd(&counter[b], 1);
    if (pos < CAP) cand[(size_t)b * CAP + pos] = key;
  }
}

// -------- per-image: async-copy candidates to LDS, bitonic sort, top-1024 state
__global__ void __launch_bounds__(1024)
sort_topk_kernel(const float* __restrict__ pred,
                 const unsigned long long* __restrict__ cand,
                 const int* __restrict__ counter,
                 float* __restrict__ topS, int* __restrict__ topLab,
                 float* __restrict__ topBox) {
  unsigned long long* sh = (unsigned long long*)dyn_smem;   // CAP * 8 = 64 KB
  int b = blockIdx.x, tid = threadIdx.x;
  int cnt = counter[b];
  if (cnt > CAP) cnt = CAP;

  // CDNA5 async memory->LDS copy (no VGPR staging), tracked by ASYNCcnt.
  // LDS byte offset = low 32 bits of the generic shared pointer (flat->LDS
  // mapping truncates to addr[31:0]).
  {
    const char* gbase = (const char*)(cand + (size_t)b * CAP);
    unsigned ldsBase = (unsigned)(uintptr_t)sh;
    for (int u = tid; u < (CAP * 8) / 16; u += 1024) {
      unsigned ldsAddr = ldsBase + (unsigned)(u * 16);
      unsigned gOff = (unsigned)(u * 16);
      asm volatile("global_load_async_to_lds_b128 %0, %1, %2"
                   :: "v"(ldsAddr), "v"(gOff), "s"(gbase) : "memory");
    }
    asm volatile("s_wait_asynccnt 0x0" ::: "memory");
  }
  __syncthreads();
  for (int i = cnt + tid; i < CAP; i += 1024) sh[i] = 0ull;  // deterministic tail
  __syncthreads();

  for (int k = 2; k <= CAP; k <<= 1) {
    for (int j = k >> 1; j > 0; j >>= 1) {
      for (int i = tid; i < CAP; i += 1024) {
        int ixj = i ^ j;
        if (ixj > i) {
          unsigned long long a = sh[i], c2 = sh[ixj];
          bool desc = ((i & k) == 0);
          if (desc ? (a < c2) : (a > c2)) { sh[i] = c2; sh[ixj] = a; }
        }
      }
      __syncthreads();
    }
  }
  int valid = cnt < K_TOP ? cnt : K_TOP;
  if (tid < K_TOP) {
    size_t o = (size_t)b * K_TOP + tid;
    if (tid < valid) {
      unsigned long long key = sh[tid];
      float s = __uint_as_float((unsigned)(key >> 32));
      unsigned idx = ~(unsigned)key;
      int n = idx / NCLS, c = idx - n * NCLS;
      const float* row = pred + ((size_t)b * NROWS + n) * ROWLEN;
      float cx = row[0], cy = row[1], w = row[2], h = row[3];
      topS[o] = s;
      topLab[o] = c;
      topBox[o * 4 + 0] = cx - 0.5f * w;
      topBox[o * 4 + 1] = cy - 0.5f * h;
      topBox[o * 4 + 2] = cx + 0.5f * w;
      topBox[o * 4 + 3] = cy + 0.5f * h;
    } else {
      topS[o] = -1.0f; topLab[o] = 0;
      topBox[o * 4 + 0] = 0.f; topBox[o * 4 + 1] = 0.f;
      topBox[o * 4 + 2] = 0.f; topBox[o * 4 + 3] = 0.f;
    }
  }
}

// -------- per-image NMS: full 1024x1024 bitmask in LDS; area sums via WMMA f32
__global__ void __launch_bounds__(1024)
nms_kernel(const float* __restrict__ topS, const int* __restrict__ topLab,
           const float* __restrict__ topBox, float* __restrict__ finalS) {
  float* x0 = (float*)dyn_smem;               // 1024 each
  float* y0 = x0 + K_TOP;
  float* x1 = y0 + K_TOP;
  float* y1 = x1 + K_TOP;
  float* ar = y1 + K_TOP;
  unsigned* mask = (unsigned*)(ar + K_TOP);   // [1024][32] = 128 KB
  unsigned* keepw = mask + K_TOP * 32;        // [32]

  int b = blockIdx.x, tid = threadIdx.x;
  size_t o = (size_t)b * K_TOP + tid;
  float s = topS[o];
  float off = (float)topLab[o] * MAX_COORD;
  float a0 = topBox[o * 4 + 0] + off, a1 = topBox[o * 4 + 1] + off;
  float a2 = topBox[o * 4 + 2] + off, a3 = topBox[o * 4 + 3] + off;
  x0[tid] = a0; y0[tid] = a1; x1[tid] = a2; y1[tid] = a3;
  ar[tid] = (a2 - a0) * (a3 - a1);
  for (int w = 0; w < 32; ++w) mask[tid * 32 + w] = 0u;
  if (tid < 32) keepw[tid] = 0u;
  __syncthreads();
  if (s > 0.0f) atomicOr(&keepw[tid >> 5], 1u << (tid & 31));

  // 64x64 grid of 16x16 tiles; upper triangle only (j>i masked per element).
  int wave = tid >> 5, lane = tid & 31;
  bool lo = lane < 16;
  int l16 = lane & 15;
  for (int t = wave; t < 64 * 64; t += 32) {
    int ti = t >> 6, tj = t & 63;
    if (tj < ti) continue;                       // wave-uniform skip, EXEC stays full
    int bi = ti * 16, bj = tj * 16;
    // A (16x4): col0 = area_i, col1 = 1 ; B (4x16): row0 = 1, row1 = area_j
    v2f A, Bm;
    A.x = lo ? ar[bi + l16] : 0.0f;
    A.y = lo ? 1.0f : 0.0f;
    Bm.x = lo ? 1.0f : 0.0f;
    Bm.y = lo ? ar[bj + l16] : 0.0f;
    v8f acc = {};
    acc = __builtin_amdgcn_wmma_f32_16x16x4_f32(false, A, false, Bm,
                                                (short)0, acc, false, false);
    int j = bj + l16;
    float jx0 = x0[j], jy0 = y0[j], jx1 = x1[j], jy1 = y1[j];
#pragma unroll
    for (int r = 0; r < 8; ++r) {
      int i = bi + r + (lo ? 0 : 8);            // D layout: vgpr r -> rows r / r+8
      float sum = acc[r];                        // area_i + area_j from matrix pipe
      float iw = fminf(x1[i], jx1) - fmaxf(x0[i], jx0);
      float ih = fminf(y1[i], jy1) - fmaxf(y0[i], jy0);
      iw = fmaxf(iw, 0.0f); ih = fmaxf(ih, 0.0f);
      float inter = iw * ih;
      // iou > T  <=>  inter > T * (union)   (union + eps strictly positive)
      float un = sum - inter + 1e-9f;
      bool bit = (j > i) && (inter > IOU_THRES * un);
      unsigned bal = (unsigned)__ballot(bit);    // wave32: [15:0]=row i_lo, [31:16]=row i_hi
      if (lane == 0) {
        int word = tj >> 1, shft = (tj & 1) * 16;
        atomicOr(&mask[(bi + r) * 32 + word], (bal & 0xFFFFu) << shft);
        atomicOr(&mask[(bi + r + 8) * 32 + word], (bal >> 16) << shft);
      }
    }
  }
  __syncthreads();

  // greedy scan: exact reference semantics (suppressed boxes never suppress)
  for (int i = 0; i < K_TOP; ++i) {
    __syncthreads();
    if ((keepw[i >> 5] >> (i & 31)) & 1u) {
      if (tid < 32) keepw[tid] &= ~mask[i * 32 + tid];
    }
  }
  __syncthreads();
  bool kept = (keepw[tid >> 5] >> (tid & 31)) & 1u;
  finalS[o] = kept ? s : -1.0f;
}

// ------------------- per-image final top-300: 1024-wide bitonic sort in LDS
__global__ void __launch_bounds__(1024)
out_kernel(const float* __restrict__ finalS, const float* __restrict__ topBox,
           const int* __restrict__ topLab, float* __restrict__ outBoxes,
           float* __restrict__ outScores, float* __restrict__ outLabels) {
  unsigned long long* sh = (unsigned long long*)dyn_smem;  // 1024 * 8 = 8 KB
  int b = blockIdx.x, tid = threadIdx.x;
  float s = finalS[(size_t)b * K_TOP + tid];
  unsigned bits = __float_as_uint(s);
  bits ^= (bits & 0x80000000u) ? 0xFFFFFFFFu : 0x80000000u;  // order-preserving
  sh[tid] = ((unsigned long long)bits << 32) | (unsigned)(~(unsigned)tid);
  __syncthreads();
  for (int k = 2; k <= K_TOP; k <<= 1) {
    for (int j = k >> 1; j > 0; j >>= 1) {
      int ixj = tid ^ j;
      if (ixj > tid) {
        unsigned long long a = sh[tid], c2 = sh[ixj];
        bool desc = ((tid & k) == 0);
        if (desc ? (a < c2) : (a > c2)) { sh[tid] = c2; sh[ixj] = a; }
      }
      __syncthreads();
    }
  }
  if (tid < DET) {
    unsigned long long key = sh[tid];
    int k = (int)(~(unsigned)key);               // original slot index
    size_t src = (size_t)b * K_TOP + k;
    float sc = finalS[src];
    size_t d = (size_t)b * DET + tid;
    outScores[d] = sc;
    outBoxes[d * 4 + 0] = topBox[src * 4 + 0];
    outBoxes[d * 4 + 1] = topBox[src * 4 + 1];
    outBoxes[d * 4 + 2] = topBox[src * 4 + 2];
    outBoxes[d * 4 + 3] = topBox[src * 4 + 3];
    outLabels[d] = (sc > 0.0f) ? (float)topLab[src] : -1.0f;
  }
}

extern "C" void kernel_launch(void* const* d_in, const int* in_sizes, int n_in,
                              void* d_out, int out_size, void* d_ws, size_t ws_size,
                              hipStream_t stream) {
  const float* pred = (const float*)d_in[0];
  char* ws = (char*)d_ws;

  // scratch layout (8-byte types first)
  unsigned long long* cand = (unsigned long long*)(ws);                 // 1 MB
  size_t off = (size_t)BATCH * CAP * 8;
  int* hist    = (int*)(ws + off);  off += (size_t)BATCH * NBINS * 4;   // 128 KB
  int* counter = (int*)(ws + off);  off += 64;
  int* cutbin  = (int*)(ws + off);  off += 64;
  float* topS  = (float*)(ws + off); off += (size_t)BATCH * K_TOP * 4;
  int* topLab  = (int*)(ws + off);   off += (size_t)BATCH * K_TOP * 4;
  float* topBox = (float*)(ws + off); off += (size_t)BATCH * K_TOP * 16;
  float* finalS = (float*)(ws + off);

  float* outBoxes  = (float*)d_out;                   // [16,300,4]
  float* outScores = outBoxes + BATCH * DET * 4;      // [16,300]
  float* outLabels = outScores + BATCH * DET;         // [16,300]

  // zero hist + counter + cutbin (contiguous after cand)
  int zeroN = BATCH * NBINS + 32;
  zero_kernel<<<(zeroN + 255) / 256, 256, 0, stream>>>((uint32_t*)hist, zeroN);

  dim3 gScore((NROWS * NCLS + 255) / 256, BATCH);
  hist_kernel<<<gScore, 256, 0, stream>>>(pred, hist);
  scan_kernel<<<BATCH, 32, 0, stream>>>(hist, cutbin);
  compact_kernel<<<gScore, 256, 0, stream>>>(pred, cutbin, cand, counter);

  sort_topk_kernel<<<BATCH, 1024, CAP * 8, stream>>>(pred, cand, counter,
                                                     topS, topLab, topBox);

  size_t nmsLds = 5 * K_TOP * 4 + (size_t)K_TOP * 32 * 4 + 32 * 4;  // ~148 KB
  nms_kernel<<<BATCH, 1024, nmsLds, stream>>>(topS, topLab, topBox, finalS);

  out_kernel<<<BATCH, 1024, K_TOP * 8, stream>>>(finalS, topBox, topLab,
                                                 outBoxes, outScores, outLabels);
}